// PointPillarScatter_61881888801254
// MI455X (gfx1250) — compile-verified
//
#include <hip/hip_runtime.h>
#include <hip/hip_bf16.h>

// PointPillarScatter for MI455X (gfx1250, wave32).
// Pure bandwidth problem: 438.8 MB output must be written once, coalesced.
// Strategy: inverse-map in workspace, then a single coalesced gather pass with
//   - global_load_async_to_lds_b128 / s_wait_asynccnt (CDNA5 async path) for map tiles
//   - ds_load_b128 readback
//   - contiguous b128 feats loads (L2-resident, 24.6 MB)
//   - non-temporal b128 output stores (don't pollute L2)

#define NYV 496
#define NXV 432
#define CFEAT 64
static constexpr int NYNX = NYV * NXV; // 214272, multiple of 4

typedef __attribute__((ext_vector_type(4))) float v4f;
typedef __attribute__((ext_vector_type(4))) int   v4i;

// ---------------- Kernel 1: map[i] = -1 for all i (vectorized) -------------
__global__ void pps_init_map(int* __restrict__ map, int n4) {
    int i = blockIdx.x * blockDim.x + threadIdx.x;
    if (i < n4) {
        v4i m = { -1, -1, -1, -1 };
        reinterpret_cast<v4i*>(map)[i] = m;
    }
}

// ---------------- Kernel 2: map[s*NYNX + y*NX + x] = pillar id -------------
__global__ void pps_scatter_map(const int* __restrict__ coords,
                                int* __restrict__ map, int P) {
    int p = blockIdx.x * blockDim.x + threadIdx.x;
    if (p < P) {
        int s = coords[4 * p + 0];
        int y = coords[4 * p + 2];
        int x = coords[4 * p + 3];
        map[(size_t)s * NYNX + (size_t)y * NXV + x] = p;
    }
}

// ---------------- Kernel 3: coalesced gather/transpose ---------------------
// Thread t owns 4 consecutive spatial positions g4..g4+3 of sample s.
// It loads 4 map entries (via async->LDS), then for each channel block of 4
// reads 4x float4 contiguous feature words and writes 4x float4 coalesced,
// non-temporal output rows. Out-of-canvas pillars (-1) produce zeros.
__global__ void __launch_bounds__(256)
pps_gather(const float* __restrict__ feats, const int* __restrict__ map,
           float* __restrict__ out, int nspatial) {
    __shared__ __align__(16) int smap[256 * 4];

    int t  = blockIdx.x * blockDim.x + threadIdx.x;
    int g4 = t * 4;                       // spatial linear index, multiple of 4
    bool active = (g4 < nspatial);
    int g4c = active ? g4 : 0;            // clamp so the async load is always safe

    // --- CDNA5 async copy: 16B of map -> our private LDS slot ---
    // Low 32 bits of a generic pointer to a __shared__ object are the LDS offset.
    unsigned lds_off = (unsigned)(uintptr_t)(&smap[threadIdx.x * 4]);
    unsigned long long gsrc = (unsigned long long)(uintptr_t)(map + g4c);
    asm volatile("global_load_async_to_lds_b128 %0, %1, off"
                 :: "v"(lds_off), "v"(gsrc) : "memory");

    // Overlap address math with the in-flight async copy.
    int s  = g4c / NYNX;
    int yx = g4c - s * NYNX;
    float* ob = out + (size_t)s * CFEAT * NYNX + yx;   // 16B aligned (yx % 4 == 0)

    asm volatile("s_wait_asynccnt 0" ::: "memory");
    v4i p4 = *reinterpret_cast<const v4i*>(&smap[threadIdx.x * 4]); // ds_load_b128

    if (!active) return;

    bool k0 = (p4.x >= 0), k1 = (p4.y >= 0), k2 = (p4.z >= 0), k3 = (p4.w >= 0);
    const float* f0 = feats + (size_t)(k0 ? p4.x : 0) * CFEAT;
    const float* f1 = feats + (size_t)(k1 ? p4.y : 0) * CFEAT;
    const float* f2 = feats + (size_t)(k2 ? p4.z : 0) * CFEAT;
    const float* f3 = feats + (size_t)(k3 ? p4.w : 0) * CFEAT;

    const v4f z4 = { 0.f, 0.f, 0.f, 0.f };
#pragma unroll 4
    for (int c0 = 0; c0 < CFEAT; c0 += 4) {
        // 4 contiguous 16B reads (feats is L2-resident: 24.6 MB << 192 MB L2)
        v4f a = *reinterpret_cast<const v4f*>(f0 + c0); if (!k0) a = z4;
        v4f b = *reinterpret_cast<const v4f*>(f1 + c0); if (!k1) b = z4;
        v4f c = *reinterpret_cast<const v4f*>(f2 + c0); if (!k2) c = z4;
        v4f d = *reinterpret_cast<const v4f*>(f3 + c0); if (!k3) d = z4;
        // register transpose -> 4 coalesced, streaming 16B stores
        v4f r0 = { a.x, b.x, c.x, d.x };
        v4f r1 = { a.y, b.y, c.y, d.y };
        v4f r2 = { a.z, b.z, c.z, d.z };
        v4f r3 = { a.w, b.w, c.w, d.w };
        __builtin_nontemporal_store(r0, reinterpret_cast<v4f*>(ob + (size_t)(c0 + 0) * NYNX));
        __builtin_nontemporal_store(r1, reinterpret_cast<v4f*>(ob + (size_t)(c0 + 1) * NYNX));
        __builtin_nontemporal_store(r2, reinterpret_cast<v4f*>(ob + (size_t)(c0 + 2) * NYNX));
        __builtin_nontemporal_store(r3, reinterpret_cast<v4f*>(ob + (size_t)(c0 + 3) * NYNX));
    }
}

// ---------------- Fallback path (only if ws_size is too small) -------------
__global__ void pps_zero_out(float* __restrict__ out, int n4) {
    int i = blockIdx.x * blockDim.x + threadIdx.x;
    if (i < n4) {
        v4f z = { 0.f, 0.f, 0.f, 0.f };
        __builtin_nontemporal_store(z, reinterpret_cast<v4f*>(out) + i);
    }
}

__global__ void pps_scatter_direct(const float* __restrict__ feats,
                                   const int* __restrict__ coords,
                                   float* __restrict__ out, int PC) {
    int idx = blockIdx.x * blockDim.x + threadIdx.x;  // (pillar, channel)
    if (idx < PC) {
        int p = idx >> 6;
        int c = idx & 63;
        int s = coords[4 * p + 0];
        int y = coords[4 * p + 2];
        int x = coords[4 * p + 3];
        out[((size_t)s * CFEAT + c) * NYNX + (size_t)y * NXV + x] = feats[idx];
    }
}

extern "C" void kernel_launch(void* const* d_in, const int* in_sizes, int n_in,
                              void* d_out, int out_size, void* d_ws, size_t ws_size,
                              hipStream_t stream) {
    const float* feats  = (const float*)d_in[0];
    const int*   coords = (const int*)d_in[1];
    float*       out    = (float*)d_out;

    const int P = in_sizes[0] / CFEAT;                  // 96000 pillars
    const int B = out_size / (CFEAT * NYNX);            // 8
    const size_t mapElems = (size_t)B * NYNX;           // 1,714,176
    const size_t needed   = mapElems * sizeof(int);     // 6.86 MB

    if (ws_size >= needed) {
        int* map = (int*)d_ws;
        const int n4 = (int)(mapElems / 4);             // NYNX % 4 == 0
        pps_init_map   <<<(n4 + 255) / 256, 256, 0, stream>>>(map, n4);
        pps_scatter_map<<<(P  + 255) / 256, 256, 0, stream>>>(coords, map, P);
        pps_gather     <<<(n4 + 255) / 256, 256, 0, stream>>>(feats, map, out,
                                                              (int)mapElems);
    } else {
        const int n4 = out_size / 4;
        pps_zero_out      <<<(n4     + 255) / 256, 256, 0, stream>>>(out, n4);
        pps_scatter_direct<<<(P * 64 + 255) / 256, 256, 0, stream>>>(feats, coords,
                                                                     out, P * 64);
    }
}